// LRCoulomb_54597624267346
// MI455X (gfx1250) — compile-verified
//
#include <hip/hip_runtime.h>

// LR Coulomb per-molecule energy:
//   e_i  = FACTOR * sum_m  q_i * q[nb[i,m]] * inv_d * (1 - fc(d))   (masked)
//   out[mol_idx[i]] += e_i
//
// Design: bandwidth-bound (205 MB HBM -> ~8.8 us floor @ 23.3 TB/s).
//  - one wave32 per atom, 4 neighbors per lane, b128 non-temporal loads
//  - q (800 KB) gathered from L2 with regular-temporal loads
//  - wave32 shfl_xor reduction -> LDS fp32 bins (ds_add_f32)
//  - sparse flush of LDS bins -> global_atomic_add_f32 (mol_idx sorted)

#define MNB     128
#define NMOL    64
#define RC      4.6f
#define FACTOR  (13.605693122994f * 0.529177210903f)

typedef __attribute__((ext_vector_type(4))) float f32x4;
typedef __attribute__((ext_vector_type(4))) int   i32x4;

__global__ void LRCoulomb_zero_out(float* __restrict__ out) {
    out[threadIdx.x] = 0.0f;
}

__global__ __launch_bounds__(1024) void LRCoulomb_kernel(
    const float* __restrict__ d_ij,
    const float* __restrict__ q,
    const int*   __restrict__ nbmat,
    const int*   __restrict__ mol_idx,
    float*       __restrict__ out,
    int n_atoms)
{
    __shared__ float bins[NMOL];
    const int tid = threadIdx.x;
    if (tid < NMOL) bins[tid] = 0.0f;
    __syncthreads();

    const int lane = tid & 31;                       // wave32 lane id
    const int wave = tid >> 5;                       // 32 waves / block
    const int atom = blockIdx.x * (blockDim.x >> 5) + wave;

    if (atom < n_atoms) {
        const size_t row = (size_t)atom * MNB;
        // Single-use streams: non-temporal b128 loads keep L2 free for q.
        const f32x4 dv = __builtin_nontemporal_load(
            reinterpret_cast<const f32x4*>(d_ij + row) + lane);
        const i32x4 nb = __builtin_nontemporal_load(
            reinterpret_cast<const i32x4*>(nbmat + row) + lane);
        const float qi = q[atom];

        float acc = 0.0f;
#pragma unroll
        for (int k = 0; k < 4; ++k) {
            const float d     = dv[k];
            const int   j     = nb[k];
            const bool  valid = (j < n_atoms);           // nb==N marks padding
            const float qj    = q[valid ? j : 0];        // hot 800 KB table in L2
            const float inv   = valid ? (1.0f / d) : 0.0f;
            // exp_cutoff: exp(1 - 1/(1 - x^2)), x = clip(d/rc, 0, 1-1e-6)
            float x = d * (1.0f / RC);
            x = fminf(fmaxf(x, 0.0f), 1.0f - 1e-6f);
            const float fc = (d < RC)
                ? __expf(1.0f - 1.0f / (1.0f - x * x))   // v_exp_f32 path
                : 0.0f;
            // e_full - e_sr contribution = qd * (1 - fc)
            acc = fmaf(qj * inv, 1.0f - fc, acc);
        }
        acc *= qi;

        // wave32 butterfly reduction (5 steps)
#pragma unroll
        for (int off = 16; off > 0; off >>= 1)
            acc += __shfl_xor(acc, off, 32);

        if (lane == 0) {
            const int mol = mol_idx[atom];               // sorted -> 1-2 bins/block
            atomicAdd(&bins[mol], FACTOR * acc);         // ds_add_f32
        }
    }
    __syncthreads();

    // Flush only the bins this block actually touched.
    if (tid < NMOL) {
        const float v = bins[tid];
        if (v != 0.0f) atomicAdd(&out[tid], v);          // global_atomic_add_f32
    }
}

extern "C" void kernel_launch(void* const* d_in, const int* in_sizes, int n_in,
                              void* d_out, int out_size, void* d_ws, size_t ws_size,
                              hipStream_t stream) {
    // setup_inputs() order: d_ij [N*M] f32, q [N] f32, nbmat [N*M] i32, mol_idx [N] i32
    const float* d_dij = (const float*)d_in[0];
    const float* d_q   = (const float*)d_in[1];
    const int*   d_nb  = (const int*)d_in[2];
    const int*   d_mol = (const int*)d_in[3];
    float* out = (float*)d_out;

    const int n_atoms = in_sizes[1];                 // N from q's element count

    // d_out is poisoned by the harness; zero it on-stream every call.
    LRCoulomb_zero_out<<<1, NMOL, 0, stream>>>(out);

    const int block = 1024;                          // 32 wave32's -> 32 atoms/block
    const int atoms_per_block = block / 32;
    const int grid = (n_atoms + atoms_per_block - 1) / atoms_per_block;
    LRCoulomb_kernel<<<grid, block, 0, stream>>>(d_dij, d_q, d_nb, d_mol, out, n_atoms);
}